// GeoScaleFormer_51470888075608
// MI455X (gfx1250) — compile-verified
//
#include <hip/hip_runtime.h>

#define BB 8
#define NN 4096
#define DD 128
#define KNB 16
#define NPT 512
#define WLK 6
#define KMAX 144   // max (K+3)&~3 across all layers (140 -> 140, pad headroom)

typedef __attribute__((ext_vector_type(2))) float v2f;
typedef __attribute__((ext_vector_type(8))) float v8f;

// ---------------------------------------------------------------------------
// Generic fp32 WMMA GEMM: Out[M,Nd] = act(A'[M,K] @ B + bias) (+residual)
//   A' row r  = A[gidx ? (r/Srows)*rowsA + gidx[r] : r]
//   B[k][n]   = transB ? W[n*K+k] : W[k*Nd+n]
// Block = 4 waves covering 4 consecutive 16-row tiles of the same 16-col tile.
// The whole B panel [Kpad x 16] is staged once into LDS (zero-padded), so the
// K loop is branch-free: global b64 A-pair + 2 conflict-free ds loads + WMMA.
// ---------------------------------------------------------------------------
template <bool RELU>
__global__ void __launch_bounds__(128)
wmma_gemm_f32(const float* __restrict__ A, const float* __restrict__ W,
              const float* __restrict__ bias, const float* __restrict__ residual,
              float* __restrict__ Out, int M, int K, int Nd, int transB,
              const int* __restrict__ gidx, int Srows, int rowsA)
{
    __shared__ float sB[KMAX * 16];

    const int tid  = threadIdx.x;
    const int lane = tid & 31;
    const int warp = tid >> 5;
    const int m0   = (blockIdx.x * 4 + warp) * 16;
    const int n0   = blockIdx.y * 16;

    const int Kpad = (K + 3) & ~3;

    // --- cooperative stage of B tile [Kpad x 16] into LDS, zero-padded ---
    for (int i = tid; i < Kpad * 16; i += 128) {
        const int k = i >> 4;
        const int n = i & 15;
        float v = 0.f;
        if (k < K && (n0 + n) < Nd)
            v = transB ? W[(size_t)(n0 + n) * K + k]
                       : W[(size_t)k * Nd + (n0 + n)];
        sB[i] = v;
    }
    __syncthreads();

    // --- per-lane A row pointer (clamped; invalid rows masked at store) ---
    const int mloc  = lane & 15;
    const int khalf = (lane >> 4) * 2;      // A/B element pair starts at this K
    const int row   = m0 + mloc;
    const int rowc  = row < M ? row : M - 1;
    long long arow  = rowc;
    if (gidx) { const int b = rowc / Srows; arow = (long long)b * rowsA + gidx[rowc]; }
    const float* __restrict__ Arow = A + (size_t)arow * (size_t)K;
    const float* __restrict__ Brow = sB + khalf * 16 + (lane & 15);

    v8f acc = {0.f, 0.f, 0.f, 0.f, 0.f, 0.f, 0.f, 0.f};

    const int K4 = K & ~3;
#pragma unroll 4
    for (int kk = 0; kk < K4; kk += 4) {
        const int ka = kk + khalf;
        v2f av = { Arow[ka], Arow[ka + 1] };
        v2f bv = { Brow[kk * 16], Brow[(kk + 1) * 16] };
        acc = __builtin_amdgcn_wmma_f32_16x16x4_f32(
                  false, av, false, bv, (short)0, acc, false, false);
    }
    if (K4 < K) {   // tail (e.g. K==3): clamped loads + cndmask zero-select
        const int ka  = K4 + khalf;
        const int kc0 = (ka     < K) ? ka     : K - 1;
        const int kc1 = (ka + 1 < K) ? ka + 1 : K - 1;
        float a0 = Arow[kc0]; if (ka     >= K) a0 = 0.f;
        float a1 = Arow[kc1]; if (ka + 1 >= K) a1 = 0.f;
        v2f av = { a0, a1 };
        v2f bv = { Brow[K4 * 16], Brow[(K4 + 1) * 16] };  // LDS zero-padded
        acc = __builtin_amdgcn_wmma_f32_16x16x4_f32(
                  false, av, false, bv, (short)0, acc, false, false);
    }

    // --- epilogue: bias / relu / residual, guarded stores only ---
    const int  col    = n0 + (lane & 15);
    const bool cvalid = col < Nd;
    const float bcol  = (bias != nullptr && cvalid) ? bias[col] : 0.f;
#pragma unroll
    for (int v = 0; v < 8; ++v) {
        const int r = m0 + v + 8 * (lane >> 4); // C/D layout: row = v + 8*(lane>>4)
        if (r < M && cvalid) {
            float val = acc[v] + bcol;
            if (RELU) val = fmaxf(val, 0.f);
            if (residual) val += residual[(size_t)r * Nd + col];
            Out[(size_t)r * Nd + col] = val;
        }
    }
}

// ---------------------------------------------------------------------------
// Brute-force kNN (k=16, includes self). One thread per query point; candidate
// points staged through LDS in 256-point tiles. Register insertion sort with
// strict '<' reproduces lax.top_k(-d) stable tie-breaking.
// ---------------------------------------------------------------------------
__global__ void __launch_bounds__(256)
knn_kernel(const float* __restrict__ xyz, int* __restrict__ knn_idx)
{
    __shared__ float sx[256], sy[256], sz[256];
    const int t   = blockIdx.x * 256 + threadIdx.x;   // query id in [0, B*N)
    const int b   = t >> 12;                           // N = 4096
    const float* X = xyz + (size_t)b * NN * 3;

    const float qx = xyz[(size_t)t * 3 + 0];
    const float qy = xyz[(size_t)t * 3 + 1];
    const float qz = xyz[(size_t)t * 3 + 2];

    float dst[KNB]; int idx[KNB];
#pragma unroll
    for (int i = 0; i < KNB; ++i) { dst[i] = 3.4e38f; idx[i] = 0; }

    for (int c0 = 0; c0 < NN; c0 += 256) {
        const int ci = c0 + threadIdx.x;
        sx[threadIdx.x] = X[(size_t)ci * 3 + 0];
        sy[threadIdx.x] = X[(size_t)ci * 3 + 1];
        sz[threadIdx.x] = X[(size_t)ci * 3 + 2];
        __syncthreads();
        for (int j = 0; j < 256; ++j) {
            const float dx = qx - sx[j], dy = qy - sy[j], dz = qz - sz[j];
            const float d  = dx * dx + dy * dy + dz * dz;
            if (d < dst[KNB - 1]) {
                dst[KNB - 1] = d; idx[KNB - 1] = c0 + j;
#pragma unroll
                for (int p = KNB - 1; p > 0; --p) {
                    if (dst[p] < dst[p - 1]) {
                        float td = dst[p]; dst[p] = dst[p - 1]; dst[p - 1] = td;
                        int   ti = idx[p]; idx[p] = idx[p - 1]; idx[p - 1] = ti;
                    }
                }
            }
        }
        __syncthreads();
    }
#pragma unroll
    for (int i = 0; i < KNB; ++i) knn_idx[(size_t)t * KNB + i] = idx[i];
}

// ---------------------------------------------------------------------------
// GHE stats: r1 = mean|e|, r2 = mean|mean(e)-e| over 16 neighbors, and
// assemble fused_in row = [hpe(128) | r1(3) | 0(3) | r2(3) | 0(3)]  (width 140)
// ---------------------------------------------------------------------------
__global__ void __launch_bounds__(256)
ghe_stats_kernel(const float* __restrict__ xyz, const int* __restrict__ knn_idx,
                 const float* __restrict__ hpe, float* __restrict__ fused_in)
{
    const int t = blockIdx.x * 256 + threadIdx.x;
    if (t >= BB * NN) return;
    const int b = t >> 12;
    const float* X = xyz + (size_t)b * NN * 3;

    const float qx = xyz[(size_t)t * 3 + 0];
    const float qy = xyz[(size_t)t * 3 + 1];
    const float qz = xyz[(size_t)t * 3 + 2];

    const int* nb = knn_idx + (size_t)t * KNB;
    float sax = 0.f, say = 0.f, saz = 0.f;   // sum |e|
    float sex = 0.f, sey = 0.f, sez = 0.f;   // sum e
#pragma unroll
    for (int j = 0; j < KNB; ++j) {
        const int n = nb[j];
        const float ex = qx - X[(size_t)n * 3 + 0];
        const float ey = qy - X[(size_t)n * 3 + 1];
        const float ez = qz - X[(size_t)n * 3 + 2];
        sax += fabsf(ex); say += fabsf(ey); saz += fabsf(ez);
        sex += ex;        sey += ey;        sez += ez;
    }
    const float mex = sex * (1.f / KNB), mey = sey * (1.f / KNB), mez = sez * (1.f / KNB);
    float s2x = 0.f, s2y = 0.f, s2z = 0.f;
#pragma unroll
    for (int j = 0; j < KNB; ++j) {
        const int n = nb[j];
        const float ex = qx - X[(size_t)n * 3 + 0];
        const float ey = qy - X[(size_t)n * 3 + 1];
        const float ez = qz - X[(size_t)n * 3 + 2];
        s2x += fabsf(mex - ex); s2y += fabsf(mey - ey); s2z += fabsf(mez - ez);
    }

    float* out = fused_in + (size_t)t * 140;
    const float4* hp4 = (const float4*)(hpe + (size_t)t * DD);
    float4* out4 = (float4*)out;
#pragma unroll
    for (int i = 0; i < DD / 4; ++i) out4[i] = hp4[i];
    out[128] = sax * (1.f / KNB); out[129] = say * (1.f / KNB); out[130] = saz * (1.f / KNB);
    out[131] = 0.f; out[132] = 0.f; out[133] = 0.f;
    out[134] = s2x * (1.f / KNB); out[135] = s2y * (1.f / KNB); out[136] = s2z * (1.f / KNB);
    out[137] = 0.f; out[138] = 0.f; out[139] = 0.f;
}

// ---------------------------------------------------------------------------
// Farthest point sampling: one block per batch; min-distance held in registers
// (16 points per thread); LDS argmax reduction with first-index tie-break
// (matches jnp.argmax). Emits the *incoming* far index per step (scan semantics).
// ---------------------------------------------------------------------------
__global__ void __launch_bounds__(256)
fps_kernel(const float* __restrict__ xyz, int* __restrict__ start_idx)
{
    const int b   = blockIdx.x;
    const int tid = threadIdx.x;
    const float* X = xyz + (size_t)b * NN * 3;

    float md[NN / 256];
#pragma unroll
    for (int j = 0; j < NN / 256; ++j) md[j] = 1e10f;

    __shared__ float sval[256];
    __shared__ int   sidx[256];

    int far = 0;
    for (int it = 0; it < NPT; ++it) {
        if (tid == 0) start_idx[b * NPT + it] = far;
        const float cx = X[(size_t)far * 3 + 0];
        const float cy = X[(size_t)far * 3 + 1];
        const float cz = X[(size_t)far * 3 + 2];

        float best = -1.f; int bi = 0x7fffffff;
#pragma unroll
        for (int j = 0; j < NN / 256; ++j) {
            const int n = tid + j * 256;
            const float dx = X[(size_t)n * 3 + 0] - cx;
            const float dy = X[(size_t)n * 3 + 1] - cy;
            const float dz = X[(size_t)n * 3 + 2] - cz;
            const float d = dx * dx + dy * dy + dz * dz;
            md[j] = fminf(md[j], d);
            if (md[j] > best || (md[j] == best && n < bi)) { best = md[j]; bi = n; }
        }
        sval[tid] = best; sidx[tid] = bi;
        __syncthreads();
        for (int s = 128; s > 0; s >>= 1) {
            if (tid < s) {
                if (sval[tid + s] > sval[tid] ||
                    (sval[tid + s] == sval[tid] && sidx[tid + s] < sidx[tid])) {
                    sval[tid] = sval[tid + s]; sidx[tid] = sidx[tid + s];
                }
            }
            __syncthreads();
        }
        far = sidx[0];
        __syncthreads();
    }
}

// ---------------------------------------------------------------------------
// One guided-walk hop: score_j = <qk_row, fused[nbr_j]>, argmax -> next index.
// qk = fused[cur] @ (Wq Wk^T) computed beforehand by the WMMA GEMM.
// ---------------------------------------------------------------------------
__global__ void __launch_bounds__(128)
walk_step_kernel(const float* __restrict__ fused, const float* __restrict__ qk,
                 const int* __restrict__ knn_idx, const int* __restrict__ cur,
                 int* __restrict__ nxt)
{
    const int t = blockIdx.x * 128 + threadIdx.x;
    if (t >= BB * NPT) return;
    const int b = t >> 9;                      // NPT = 512
    const int c = cur[t];
    const int* nb = knn_idx + ((size_t)b * NN + c) * KNB;
    const float4* q4 = (const float4*)(qk + (size_t)t * DD);

    float best = -3.4e38f; int bestn = nb[0];
#pragma unroll 1
    for (int j = 0; j < KNB; ++j) {
        const int n = nb[j];
        const float4* f4 = (const float4*)(fused + ((size_t)b * NN + n) * DD);
        float s = 0.f;
#pragma unroll
        for (int i = 0; i < DD / 4; ++i) {
            const float4 a = q4[i], f = f4[i];
            s += a.x * f.x + a.y * f.y + a.z * f.z + a.w * f.w;
        }
        if (s > best) { best = s; bestn = n; }  // strict '>' == argmax first-max
    }
    nxt[t] = bestn;
}

// ---------------------------------------------------------------------------
// all_points = concat(xyz, xyz[sel]+disp); coarse also written to d_out.
// ---------------------------------------------------------------------------
__global__ void __launch_bounds__(256)
assemble_points_kernel(const float* __restrict__ xyz, const int* __restrict__ sel,
                       const float* __restrict__ disp, float* __restrict__ all_points,
                       float* __restrict__ out_coarse)
{
    const int t = blockIdx.x * 256 + threadIdx.x;
    const int total = BB * (NN + NPT);
    if (t >= total) return;
    const int b = t / (NN + NPT);
    const int l = t - b * (NN + NPT);
    if (l < NN) {
#pragma unroll
        for (int d = 0; d < 3; ++d)
            all_points[(size_t)t * 3 + d] = xyz[((size_t)b * NN + l) * 3 + d];
    } else {
        const int s = l - NN;
        const int g = b * NPT + s;
        const int si = sel[g];
#pragma unroll
        for (int d = 0; d < 3; ++d) {
            const float v = xyz[((size_t)b * NN + si) * 3 + d] + disp[(size_t)g * 3 + d];
            all_points[(size_t)t * 3 + d] = v;
            out_coarse[(size_t)g * 3 + d] = v;
        }
    }
}

// ---------------------------------------------------------------------------
static void launch_gemm(hipStream_t st, const float* A, const float* W,
                        const float* bias, const float* resid, float* Out,
                        int M, int K, int Nd, bool relu, int transB,
                        const int* gidx, int Srows, int rowsA)
{
    dim3 grid((M + 63) / 64, (Nd + 15) / 16);
    if (relu)
        wmma_gemm_f32<true><<<grid, 128, 0, st>>>(A, W, bias, resid, Out, M, K, Nd,
                                                  transB, gidx, Srows, rowsA);
    else
        wmma_gemm_f32<false><<<grid, 128, 0, st>>>(A, W, bias, resid, Out, M, K, Nd,
                                                   transB, gidx, Srows, rowsA);
}

extern "C" void kernel_launch(void* const* d_in, const int* in_sizes, int n_in,
                              void* d_out, int out_size, void* d_ws, size_t ws_size,
                              hipStream_t stream)
{
    (void)in_sizes; (void)n_in; (void)out_size; (void)ws_size;

    const float* xyz   = (const float*)d_in[0];
    const float* encW1 = (const float*)d_in[1];
    const float* encb1 = (const float*)d_in[2];
    const float* encW2 = (const float*)d_in[3];
    const float* encb2 = (const float*)d_in[4];
    const float* prjW1 = (const float*)d_in[5];
    const float* prjb1 = (const float*)d_in[6];
    const float* prjW2 = (const float*)d_in[7];
    const float* prjb2 = (const float*)d_in[8];
    const float* fusW1 = (const float*)d_in[9];
    const float* fusb1 = (const float*)d_in[10];
    const float* fusW2 = (const float*)d_in[11];
    const float* fusb2 = (const float*)d_in[12];
    const float* wlkWq = (const float*)d_in[13];
    const float* wlkWk = (const float*)d_in[14];
    const float* wlkWd1= (const float*)d_in[15];
    const float* wlkbd1= (const float*)d_in[16];
    const float* wlkWd2= (const float*)d_in[17];
    const float* wlkbd2= (const float*)d_in[18];
    const float* refW1 = (const float*)d_in[19];
    const float* refb1 = (const float*)d_in[20];
    const float* refW2 = (const float*)d_in[21];
    const float* refb2 = (const float*)d_in[22];

    char* ws = (char*)d_ws;
    size_t off = 0;
    auto alloc = [&](size_t nbytes) -> void* {
        off = (off + 255) & ~(size_t)255;
        void* p = ws + off;
        off += nbytes;
        return p;
    };

    const int M  = BB * NN;               // 32768
    const int M2 = BB * (NN + NPT);       // 36864
    const int MS = BB * NPT;              // 4096

    float* feats  = (float*)alloc((size_t)M  * DD  * 4);
    float* ffeats = (float*)alloc((size_t)M  * DD  * 4);
    float* scr1   = (float*)alloc((size_t)M2 * DD  * 4);   // reused hidden buffers
    float* scr2   = (float*)alloc((size_t)M2 * DD  * 4);
    float* fin    = (float*)alloc((size_t)M  * 140 * 4);
    int*   knnidx = (int*)  alloc((size_t)M  * KNB * 4);
    int*   cur0   = (int*)  alloc((size_t)MS * 4);
    int*   cur1   = (int*)  alloc((size_t)MS * 4);
    float* Wqk    = (float*)alloc((size_t)DD * DD * 4);
    float* qk     = (float*)alloc((size_t)MS * DD * 4);    // also reused for sel hidden
    float* disp   = (float*)alloc((size_t)MS * 3  * 4);
    float* allp   = (float*)alloc((size_t)M2 * 3  * 4);

    float* out_coarse = (float*)d_out;               // [B,512,3]
    float* out_fine   = (float*)d_out + (size_t)MS * 3;

    // --- encoder: feats = relu(xyz@W1+b1)@W2+b2 ---
    launch_gemm(stream, xyz,  encW1, encb1, nullptr, scr1,  M, 3,  64,  true,  0, nullptr, 0, 0);
    launch_gemm(stream, scr1, encW2, encb2, nullptr, feats, M, 64, DD,  false, 0, nullptr, 0, 0);

    // --- hpe = relu(xyz@prjW1+b1)@prjW2+b2 ---
    launch_gemm(stream, xyz,  prjW1, prjb1, nullptr, scr1, M, 3,  DD, true,  0, nullptr, 0, 0);
    launch_gemm(stream, scr1, prjW2, prjb2, nullptr, scr2, M, DD, DD, false, 0, nullptr, 0, 0);

    // --- kNN + GHE stats -> fused_in [M,140] ---
    knn_kernel<<<M / 256, 256, 0, stream>>>(xyz, knnidx);
    ghe_stats_kernel<<<(M + 255) / 256, 256, 0, stream>>>(xyz, knnidx, scr2, fin);

    // --- pos_embed + residual: fused_feats = feats + mlp2(fused_in) ---
    launch_gemm(stream, fin,  fusW1, fusb1, nullptr, scr1,   M, 140, DD, true,  0, nullptr, 0, 0);
    launch_gemm(stream, scr1, fusW2, fusb2, feats,   ffeats, M, DD,  DD, false, 0, nullptr, 0, 0);

    // --- FPS start indices ---
    fps_kernel<<<BB, 256, 0, stream>>>(xyz, cur0);

    // --- Wqk = Wq @ Wk^T (hop invariant) ---
    launch_gemm(stream, wlkWq, wlkWk, nullptr, nullptr, Wqk, DD, DD, DD, false, 1, nullptr, 0, 0);

    // --- guided walk: 6 hops ---
    int* ca = cur0;
    int* cb = cur1;
    for (int hop = 0; hop < WLK; ++hop) {
        launch_gemm(stream, ffeats, Wqk, nullptr, nullptr, qk,
                    MS, DD, DD, false, 0, ca, NPT, NN);
        walk_step_kernel<<<(MS + 127) / 128, 128, 0, stream>>>(ffeats, qk, knnidx, ca, cb);
        int* tmp = ca; ca = cb; cb = tmp;
    }
    // final indices in `ca`

    // --- coarse displacement MLP on gathered features ---
    launch_gemm(stream, ffeats, wlkWd1, wlkbd1, nullptr, qk,   MS, DD, 64, true,  0, ca, NPT, NN);
    launch_gemm(stream, qk,     wlkWd2, wlkbd2, nullptr, disp, MS, 64, 3,  false, 0, nullptr, 0, 0);

    // --- assemble all_points and write coarse output ---
    assemble_points_kernel<<<(M2 + 255) / 256, 256, 0, stream>>>(xyz, ca, disp, allp, out_coarse);

    // --- refine: all_feats = enc(all_points); fine = mlp2(all_feats, ref) ---
    launch_gemm(stream, allp, encW1, encb1, nullptr, scr1,     M2, 3,  64, true,  0, nullptr, 0, 0);
    launch_gemm(stream, scr1, encW2, encb2, nullptr, scr2,     M2, 64, DD, false, 0, nullptr, 0, 0);
    launch_gemm(stream, scr2, refW1, refb1, nullptr, scr1,     M2, DD, DD, true,  0, nullptr, 0, 0);
    launch_gemm(stream, scr1, refW2, refb2, nullptr, out_fine, M2, DD, 3,  false, 0, nullptr, 0, 0);
}